// PointSetEmbedding_49271864819780
// MI455X (gfx1250) — compile-verified
//
#include <hip/hip_runtime.h>
#include <math.h>

#define NPTS   16384
#define NPOINT 1024
#define NSAMP  32
#define BATCH  8
#define NCOL   32768   // NPOINT * NSAMP columns per batch
#define C0     68      // 3 + 64 input channels, padded 67 -> 68 (mult of 4)
#define C1     64
#define C2     128

typedef __attribute__((ext_vector_type(2))) float v2f;
typedef __attribute__((ext_vector_type(8))) float v8f;

// ---------------------------------------------------------------------------
// Phase 1: farthest point sampling. One 1024-thread block per batch; each
// thread owns 16 points (stride 1024) in registers. 1024 sequential steps,
// each: broadcast centroid, min-dist update, argmax (wave shuffle + LDS).
// Tie-break matches jnp.argmax (first occurrence = smallest index).
// ---------------------------------------------------------------------------
__global__ __launch_bounds__(1024) void fps_kernel(const float* __restrict__ xyz,
                                                   int* __restrict__ fps_idx) {
  const int b = blockIdx.x;
  const int t = threadIdx.x;
  const float* xb = xyz + (size_t)b * 3 * NPTS;
  float px[16], py[16], pz[16], dist[16];
#pragma unroll
  for (int i = 0; i < 16; ++i) {
    int n = t + i * 1024;
    px[i] = xb[n];
    py[i] = xb[NPTS + n];
    pz[i] = xb[2 * NPTS + n];
    dist[i] = __builtin_inff();
  }
  __shared__ float swv[32];
  __shared__ int   swi[32];
  __shared__ float scen[3];
  __shared__ int   scur;
  if (t == 0) scur = 0;
  __syncthreads();
  for (int s = 0; s < NPOINT; ++s) {
    int f = scur;
    if (t == (f & 1023)) {
      int slot = f >> 10;
      scen[0] = px[slot]; scen[1] = py[slot]; scen[2] = pz[slot];
    }
    if (t == 0) fps_idx[b * NPOINT + s] = f;
    __syncthreads();
    float cx = scen[0], cy = scen[1], cz = scen[2];
    float bv = -1.f; int bi = 0;
#pragma unroll
    for (int i = 0; i < 16; ++i) {
      float dx = px[i] - cx, dy = py[i] - cy, dz = pz[i] - cz;
      float d = dx * dx + dy * dy + dz * dz;
      float nd = fminf(dist[i], d);
      dist[i] = nd;
      if (nd > bv) { bv = nd; bi = t + i * 1024; }   // ascending index order
    }
    // wave32 reduce: (max value, then min index on tie)
#pragma unroll
    for (int m = 1; m < 32; m <<= 1) {
      float ov = __shfl_xor(bv, m, 32);
      int   oi = __shfl_xor(bi, m, 32);
      if (ov > bv || (ov == bv && oi < bi)) { bv = ov; bi = oi; }
    }
    if ((t & 31) == 0) { swv[t >> 5] = bv; swi[t >> 5] = bi; }
    __syncthreads();
    if (t < 32) {
      bv = swv[t]; bi = swi[t];
#pragma unroll
      for (int m = 1; m < 32; m <<= 1) {
        float ov = __shfl_xor(bv, m, 32);
        int   oi = __shfl_xor(bi, m, 32);
        if (ov > bv || (ov == bv && oi < bi)) { bv = ov; bi = oi; }
      }
      if (t == 0) scur = bi;
    }
    __syncthreads();
  }
}

// ---------------------------------------------------------------------------
// Phase 2: ball query (first 32 in-radius indices, ascending) + gather.
// One wave32 per (batch, center): lane j ends holding slot j's source index
// via ballot + rank binary-search; then 68 coalesced channel writes.
// ---------------------------------------------------------------------------
__global__ __launch_bounds__(128) void group_kernel(const float* __restrict__ xyz,
                                                    const float* __restrict__ points,
                                                    const int* __restrict__ fps_idx,
                                                    float* __restrict__ X0) {
  const float R2 = (float)(0.2 * 0.2);
  int wave = (int)(blockIdx.x * blockDim.x + threadIdx.x) >> 5;
  int lane = threadIdx.x & 31;
  int b = wave >> 10;
  int s = wave & 1023;
  const float* xb = xyz + (size_t)b * 3 * NPTS;
  int f = fps_idx[b * NPOINT + s];
  float nx = xb[f], ny = xb[NPTS + f], nz = xb[2 * NPTS + f];
  int sel = -1;
  int count = 0;
  for (int n0 = 0; n0 < NPTS && count < NSAMP; n0 += 32) {
    int n = n0 + lane;
    float dx = xb[n] - nx, dy = xb[NPTS + n] - ny, dz = xb[2 * NPTS + n] - nz;
    float sq = dx * dx + dy * dy + dz * dz;
    unsigned mask = (unsigned)__ballot(sq <= R2);
    int cnt = __popc(mask);
    int need = lane - count;            // my slot's rank within this chunk
    if (need >= 0 && need < cnt) {      // find the need-th set bit of mask
      unsigned m = mask; int p = 0, c;
      c = __popc(m & 0xFFFFu); if (need >= c) { need -= c; p += 16; m >>= 16; }
      c = __popc(m & 0x00FFu); if (need >= c) { need -= c; p += 8;  m >>= 8; }
      c = __popc(m & 0x000Fu); if (need >= c) { need -= c; p += 4;  m >>= 4; }
      c = __popc(m & 0x0003u); if (need >= c) { need -= c; p += 2;  m >>= 2; }
      c = __popc(m & 0x0001u); if (need >= c) {            p += 1;           }
      sel = n0 + p;
    }
    count += cnt;
  }
  int first = __shfl(sel, 0, 32);       // center itself always in radius
  if (sel < 0) sel = first;             // pad short groups with first index
  // X0 layout: [b][c][s*32 + k], c = 0..2 centered xyz, 3..66 points, 67 zero
  float* xo = X0 + (size_t)b * C0 * NCOL + s * NSAMP + lane;
  xo[0 * (size_t)NCOL] = xb[sel] - nx;
  xo[1 * (size_t)NCOL] = xb[NPTS + sel] - ny;
  xo[2 * (size_t)NCOL] = xb[2 * NPTS + sel] - nz;
  const float* pb = points + (size_t)b * 64 * NPTS;
#pragma unroll 4
  for (int c = 0; c < 64; ++c)
    xo[(size_t)(3 + c) * NCOL] = pb[(size_t)c * NPTS + sel];
  xo[(size_t)67 * NCOL] = 0.f;
}

// ---------------------------------------------------------------------------
// Phase 3/4: fp32 GEMM  Y = silu(W * X + bias)  via V_WMMA_F32_16X16X4_F32.
// Block = 256 threads = 8 waves arranged 4(M) x 2(N): 64x32 output tile.
// A(16x4): lane (h=lane>>4, m=lane&15) holds W[m0+m][k+2h], W[m0+m][k+2h+1].
// B(4x16): lane holds X[k+2h][n0+n], X[k+2h+1][n0+n].
// FUSE_MEAN: block's 32 columns == one center's k-group -> reduce to output.
// ---------------------------------------------------------------------------
template <bool FUSE_MEAN>
__global__ __launch_bounds__(256) void gemm_kernel(const float* __restrict__ W,
                                                   const float* __restrict__ bias,
                                                   const float* __restrict__ X,
                                                   float* __restrict__ Y,
                                                   int M, int Kdim) {
  const int b    = blockIdx.z;
  const int wave = threadIdx.x >> 5;
  const int lane = threadIdx.x & 31;
  const int wr   = wave >> 1;
  const int wc   = wave & 1;
  const int m0   = blockIdx.y * 64 + wr * 16;
  const int n0   = blockIdx.x * 32 + wc * 16;
  const int ml   = lane & 15;
  const int h    = lane >> 4;
  const float* Xb = X + (size_t)b * Kdim * NCOL;
  v8f acc = {0.f, 0.f, 0.f, 0.f, 0.f, 0.f, 0.f, 0.f};
  const float* wp = W + (size_t)(m0 + ml) * Kdim + 2 * h;
  const float* xp = Xb + (size_t)(2 * h) * NCOL + n0 + ml;
  for (int kk = 0; kk < Kdim; kk += 4) {
    v2f a;  a.x  = wp[0]; a.y  = wp[1];
    v2f bm; bm.x = xp[0]; bm.y = xp[NCOL];
    acc = __builtin_amdgcn_wmma_f32_16x16x4_f32(false, a, false, bm,
                                                (short)0, acc, false, false);
    wp += 4;
    xp += (size_t)4 * NCOL;
  }
  if constexpr (!FUSE_MEAN) {
    float* yb = Y + ((size_t)b * M + m0 + 8 * h) * NCOL + n0 + ml;
#pragma unroll
    for (int r = 0; r < 8; ++r) {
      int row = m0 + r + 8 * h;
      float v = acc[r] + bias[row];
      v = v / (1.f + expf(-v));
      yb[(size_t)r * NCOL] = v;
    }
  } else {
    __shared__ float accs[64];
    if (threadIdx.x < 64) accs[threadIdx.x] = 0.f;
    __syncthreads();
#pragma unroll
    for (int r = 0; r < 8; ++r) {
      int row = m0 + r + 8 * h;
      float v = acc[r] + bias[row];
      v = v / (1.f + expf(-v));
      v += __shfl_xor(v, 1, 32);        // sum over 16 cols within lane-half
      v += __shfl_xor(v, 2, 32);
      v += __shfl_xor(v, 4, 32);
      v += __shfl_xor(v, 8, 32);
      if (ml == 0) atomicAdd(&accs[wr * 16 + r + 8 * h], v);  // ds_add_f32
    }
    __syncthreads();
    if (threadIdx.x < 64) {
      int row = blockIdx.y * 64 + threadIdx.x;
      Y[((size_t)b * 256 + row) * NPOINT + blockIdx.x] =
          accs[threadIdx.x] * (1.f / 32.f);
    }
  }
}

__global__ void pad_w0_kernel(const float* __restrict__ w0, float* __restrict__ w0p) {
  int i = blockIdx.x * blockDim.x + threadIdx.x;
  if (i < 64 * C0) {
    int m = i / C0, k = i % C0;
    w0p[i] = (k < 67) ? w0[m * 67 + k] : 0.f;
  }
}

extern "C" void kernel_launch(void* const* d_in, const int* in_sizes, int n_in,
                              void* d_out, int out_size, void* d_ws, size_t ws_size,
                              hipStream_t stream) {
  (void)in_sizes; (void)n_in; (void)out_size; (void)ws_size;
  const float* xyz    = (const float*)d_in[0];
  const float* points = (const float*)d_in[1];
  const float* w0     = (const float*)d_in[2];
  const float* b0     = (const float*)d_in[3];
  const float* w1     = (const float*)d_in[4];
  const float* b1     = (const float*)d_in[5];
  const float* w2     = (const float*)d_in[6];
  const float* b2     = (const float*)d_in[7];

  char* ws  = (char*)d_ws;
  float* w0p = (float*)ws;                       // 64*68 floats
  int*   fps = (int*)(ws + 32768);               // 8192 ints
  float* X0  = (float*)(ws + 65536);             // 8*68*32768 floats
  float* X1  = X0 + (size_t)BATCH * C0 * NCOL;   // 8*64*32768 floats
  float* X2  = X1 + (size_t)BATCH * C1 * NCOL;   // 8*128*32768 floats
  float* out = (float*)d_out;                    // (8, 256, 1024)

  fps_kernel<<<BATCH, 1024, 0, stream>>>(xyz, fps);
  pad_w0_kernel<<<(64 * C0 + 255) / 256, 256, 0, stream>>>(w0, w0p);
  group_kernel<<<(BATCH * NPOINT) / 4, 128, 0, stream>>>(xyz, points, fps, X0);
  gemm_kernel<false><<<dim3(NCOL / 32, 1, BATCH), 256, 0, stream>>>(w0p, b0, X0, X1, 64,  C0);
  gemm_kernel<false><<<dim3(NCOL / 32, 2, BATCH), 256, 0, stream>>>(w1,  b1, X1, X2, 128, C1);
  gemm_kernel<true ><<<dim3(NCOL / 32, 4, BATCH), 256, 0, stream>>>(w2,  b2, X2, out, 256, C2);
}